// SceneGraphPointNet_53412213293217
// MI455X (gfx1250) — compile-verified
//
#include <hip/hip_runtime.h>

// SceneGraphPointNet forward for MI455X (gfx1250).
// ~1.8 TFLOP fp32, ~0.4 GB traffic -> compute-bound; all linear layers run on
// V_WMMA_F32_16X16X4_F32 (native fp32 matrix pipe, exact fp32 accumulation).
// Guard-free WMMA inner loops (padded weights prepped in workspace); each wave
// computes two 16x16 row tiles sharing one B fragment. Contiguous 64KB
// global->LDS staging uses the Tensor Data Mover (tensor_load_to_lds +
// s_wait_tensorcnt) so VALU/VMEM stay free for the matrix pipe.

typedef float v2f __attribute__((ext_vector_type(2)));
typedef float v8f __attribute__((ext_vector_type(8)));
typedef unsigned int v4u __attribute__((ext_vector_type(4)));
typedef int v4i __attribute__((ext_vector_type(4)));
typedef int v8i __attribute__((ext_vector_type(8)));

static constexpr int Bc = 32, Nc = 52, Pc = 1024, Dc = 512, OUTc = 607;

// ---------------------------------------------------------------------------
// TDM: async copy of `dwords` consecutive DWORDs from global to LDS.
// D# built per CDNA5 ISA 8.3/8.4: group0 = {count=1 | lds_addr | global_addr |
// type=2}, group1 = {data_size=4B, tensor_dim0=tile_dim0=dwords, dims1=1,
// stride0=dwords}. Groups 2/3 zero (<=2D tensor). Issue from one wave only.
// ---------------------------------------------------------------------------
__device__ inline void tdm_load_lds(unsigned ldsByteOff, const void* gsrc, unsigned dwords)
{
    const unsigned long long ga = (unsigned long long)gsrc;
    v4u g0;
    g0.x = 0x1u;                                            // count=1, user D#
    g0.y = __builtin_amdgcn_groupstaticsize() + ldsByteOff; // lds_addr (bytes)
    g0.z = (unsigned)(ga & 0xffffffffu);                    // global_addr[31:0]
    g0.w = (unsigned)((ga >> 32) & 0x1ffffffu) | (2u << 30);// [56:32] | type=2
    v8i g1;
    g1[0] = 0x20000;                                 // data_size=4B, no mcast
    g1[1] = (int)((dwords & 0xffffu) << 16);         // tensor_dim0[15:0]
    g1[2] = (int)((dwords >> 16) | (1u << 16));      // tensor_dim0[31:16], dim1=1
    g1[3] = (int)((dwords & 0xffffu) << 16);         // tile_dim0
    g1[4] = 1;                                       // tile_dim1=1
    g1[5] = (int)dwords;                             // tensor_dim0_stride[31:0]
    g1[6] = 0;
    g1[7] = 0;
    v4i z4 = {};
#if defined(__clang_major__) && (__clang_major__ >= 23)
    v8i z8 = {};
    __builtin_amdgcn_tensor_load_to_lds(g0, g1, z4, z4, z8, 0);
#else
    __builtin_amdgcn_tensor_load_to_lds(g0, g1, z4, z4, 0);
#endif
}

// ---------------------------------------------------------------------------
// Two 16x16 output tiles (rows 0-15 and 16-31) of D += A(32xK) * W(KxCout).
// A from LDS, W from global (unguarded). A layout (32-bit 16x4): lanes 0-15
// hold K+0/K+1 in v0/v1, lanes 16-31 hold K+2/K+3. B mirrors over N.
// C/D: vgpr v holds row m = v + 8*(lane>=16), col n = nBase + (lane&15).
// ---------------------------------------------------------------------------
__device__ inline void wmma_tile_pair(const float* __restrict__ lin, int ldIn,
                                      const float* __restrict__ W, int wStride,
                                      int K, int nBase, int lane,
                                      v8f& acc0, v8f& acc1)
{
    const int half = lane >> 4;
    const int l    = lane & 15;
    const int kOff = half * 2;
    const float* aRow0 = lin + l * ldIn + kOff;
    const float* aRow1 = lin + (16 + l) * ldIn + kOff;
    const float* wCol  = W + nBase + l;
#pragma unroll 4
    for (int kk = 0; kk < K; kk += 4) {
        v2f a0 = *(const v2f*)(aRow0 + kk);          // ds_load_b64
        v2f a1 = *(const v2f*)(aRow1 + kk);
        const int k0 = kk + kOff;
        v2f b;
        b.x = wCol[(size_t)(k0    ) * wStride];
        b.y = wCol[(size_t)(k0 + 1) * wStride];
        acc0 = __builtin_amdgcn_wmma_f32_16x16x4_f32(false, a0, false, b,
                                                     (short)0, acc0, false, false);
        acc1 = __builtin_amdgcn_wmma_f32_16x16x4_f32(false, a1, false, b,
                                                     (short)0, acc1, false, false);
    }
}

__device__ inline void epi_pair_lds(v8f acc0, v8f acc1, float* lout, int ldOut,
                                    const float* __restrict__ bias,
                                    int nBase, int lane, bool doRelu)
{
    const int half = lane >> 4, l = lane & 15, n = nBase + l;
    const float bn = bias[n];
#pragma unroll
    for (int v = 0; v < 8; ++v) {
        const int m = v + half * 8;
        float v0 = acc0[v] + bn;
        float v1 = acc1[v] + bn;
        if (doRelu) { v0 = fmaxf(v0, 0.f); v1 = fmaxf(v1, 0.f); }
        lout[(m     ) * ldOut + n] = v0;
        lout[(m + 16) * ldOut + n] = v1;
    }
}

// Full layer: in (32 x K, LDS) -> out (32 x Cout, LDS). Cout multiple of 16.
// 8 waves per WG split the Cout/16 column tiles; each wave does 2 row tiles.
__device__ inline void layer32(const float* lin, int ldIn, float* lout, int ldOut,
                               const float* __restrict__ W, const float* __restrict__ bias,
                               int K, int Cout, bool doRelu, int wave, int lane)
{
    for (int t = wave; t < (Cout >> 4); t += 8) {
        v8f acc0 = {}, acc1 = {};
        wmma_tile_pair(lin, ldIn, W, Cout, K, t * 16, lane, acc0, acc1);
        epi_pair_lds(acc0, acc1, lout, ldOut, bias, t * 16, lane, doRelu);
    }
}

// ---------------------------------------------------------------------------
// Weight prep: zero-padded copies so all hot loops are guard-free.
// ---------------------------------------------------------------------------
__global__ void __launch_bounds__(256) prep_weights_kernel(
    const float* __restrict__ deW0, const float* __restrict__ seW0,
    const float* __restrict__ omW2, const float* __restrict__ omb2,
    float* __restrict__ deW0p, float* __restrict__ seW0p,
    float* __restrict__ omW2p, float* __restrict__ omb2p)
{
    const int tid = blockIdx.x * 256 + threadIdx.x;
    const int nthr = gridDim.x * 256;
    if (tid < 512) { const int k = tid >> 6, n = tid & 63; deW0p[tid] = (k < 6) ? deW0[k * 64 + n] : 0.f; }
    if (tid < 256) { const int k = tid >> 5, n = tid & 31; seW0p[tid] = (k < 6) ? seW0[k * 32 + n] : 0.f; }
    if (tid < 608) omb2p[tid] = (tid < OUTc) ? omb2[tid] : 0.f;
    for (int idx = tid; idx < 256 * 608; idx += nthr) {
        const int k = idx / 608, n = idx - k * 608;
        omW2p[idx] = (n < OUTc) ? omW2[k * OUTc + n] : 0.f;
    }
}

// ---------------------------------------------------------------------------
// Dense PointNet encoder: 6->64->128->256->512->512 (ReLU each), max over P.
// One WG handles 128 points of one (b,n) cloud as 4 sequential 32-row tiles.
// ---------------------------------------------------------------------------
__global__ void __launch_bounds__(256) dense_encoder_kernel(
    const float* __restrict__ objects,
    const float* __restrict__ W0, const float* __restrict__ b0,
    const float* __restrict__ W1, const float* __restrict__ b1,
    const float* __restrict__ W2, const float* __restrict__ b2,
    const float* __restrict__ W3, const float* __restrict__ b3,
    const float* __restrict__ W4, const float* __restrict__ b4,
    float* __restrict__ objFeat)
{
    extern __shared__ float smem[];
    float* bufA = smem;                 // 32*512
    float* bufB = smem + 32 * 512;      // 32*512
    float* cmax = smem + 64 * 512;      // 512

    const int wg = blockIdx.x;
    const int pc = wg & 7;
    const int n  = (wg >> 3) % Nc;
    const int b  = wg / (8 * Nc);
    const int tid = threadIdx.x, lane = tid & 31, wave = tid >> 5;

    for (int c = tid; c < Dc; c += 256) cmax[c] = 0.f;
    __syncthreads();

    const size_t objBase = ((size_t)b * Nc + n) * Pc * 6;
    for (int s = 0; s < 4; ++s) {
        const int p0 = pc * 128 + s * 32;
        {                                           // stage 32x8 (pad K 6->8)
            const int r = tid >> 3, c = tid & 7;
            bufA[tid] = (c < 6) ? objects[objBase + (size_t)(p0 + r) * 6 + c] : 0.f;
        }
        __syncthreads();
        layer32(bufA,   8, bufB,  64, W0, b0,   8,  64, true, wave, lane); __syncthreads();
        layer32(bufB,  64, bufA, 128, W1, b1,  64, 128, true, wave, lane); __syncthreads();
        layer32(bufA, 128, bufB, 256, W2, b2, 128, 256, true, wave, lane); __syncthreads();
        layer32(bufB, 256, bufA, 512, W3, b3, 256, 512, true, wave, lane); __syncthreads();
        layer32(bufA, 512, bufB, 512, W4, b4, 512, 512, true, wave, lane); __syncthreads();
        for (int c = tid; c < Dc; c += 256) {       // running column max
            float m = cmax[c];
#pragma unroll
            for (int r = 0; r < 32; ++r) m = fmaxf(m, bufB[r * 512 + c]);
            cmax[c] = m;
        }
        __syncthreads();
    }
    // Values >= 0 (ReLU) and destination zero-initialized, so integer-bit
    // atomicMax is exact for non-negative IEEE floats.
    float* dst = objFeat + ((size_t)n * Bc + b) * Dc;
    for (int c = tid; c < Dc; c += 256)
        atomicMax((int*)(dst + c), __float_as_int(cmax[c]));
}

// ---------------------------------------------------------------------------
// Sparse (relation) encoder: xyz-masked input, 6->32->64->512, max over P.
// ---------------------------------------------------------------------------
__global__ void __launch_bounds__(256) sparse_encoder_kernel(
    const float* __restrict__ objects,
    const float* __restrict__ W0, const float* __restrict__ b0,
    const float* __restrict__ W1, const float* __restrict__ b1,
    const float* __restrict__ W2, const float* __restrict__ b2,
    float* __restrict__ pnObj)
{
    extern __shared__ float smem[];
    float* bufA = smem;
    float* bufB = smem + 32 * 512;
    float* cmax = smem + 64 * 512;

    const int wg = blockIdx.x;
    const int pc = wg & 7;
    const int n  = (wg >> 3) % Nc;
    const int b  = wg / (8 * Nc);
    const int tid = threadIdx.x, lane = tid & 31, wave = tid >> 5;

    for (int c = tid; c < Dc; c += 256) cmax[c] = 0.f;
    __syncthreads();

    const size_t objBase = ((size_t)b * Nc + n) * Pc * 6;
    for (int s = 0; s < 4; ++s) {
        const int p0 = pc * 128 + s * 32;
        {                                           // keep xyz, zero the rest
            const int r = tid >> 3, c = tid & 7;
            bufA[tid] = (c < 3) ? objects[objBase + (size_t)(p0 + r) * 6 + c] : 0.f;
        }
        __syncthreads();
        layer32(bufA,  8, bufB,  32, W0, b0,  8,  32, true, wave, lane); __syncthreads();
        layer32(bufB, 32, bufA,  64, W1, b1, 32,  64, true, wave, lane); __syncthreads();
        layer32(bufA, 64, bufB, 512, W2, b2, 64, 512, true, wave, lane); __syncthreads();
        for (int c = tid; c < Dc; c += 256) {
            float m = cmax[c];
#pragma unroll
            for (int r = 0; r < 32; ++r) m = fmaxf(m, bufB[r * 512 + c]);
            cmax[c] = m;
        }
        __syncthreads();
    }
    float* dst = pnObj + ((size_t)n * Bc + b) * Dc;
    for (int c = tid; c < Dc; c += 256)
        atomicMax((int*)(dst + c), __float_as_int(cmax[c]));
}

// ---------------------------------------------------------------------------
// Object MLP on pooled dense features: 512->128->256->607 (last has no ReLU).
// 52 WGs x 32 rows; input tile staged by the Tensor Data Mover.
// ---------------------------------------------------------------------------
__global__ void __launch_bounds__(256) object_mlp_kernel(
    const float* __restrict__ objFeat,
    const float* __restrict__ W0, const float* __restrict__ b0,
    const float* __restrict__ W1, const float* __restrict__ b1,
    const float* __restrict__ W2p, const float* __restrict__ b2p,
    float* __restrict__ outObj)
{
    extern __shared__ float smem[];
    float* bufA = smem;                 // 32*512 (stage, then 32*256 L1 out)
    float* bufB = smem + 32 * 512;      // 32*128

    const int r0 = blockIdx.x * 32;
    const int tid = threadIdx.x, lane = tid & 31, wave = tid >> 5;
    const int half = lane >> 4, l = lane & 15;

    if (wave == 0) {                    // async 64KB contiguous stage via TDM
        tdm_load_lds(0u, objFeat + (size_t)r0 * Dc, 32 * 512);
        __builtin_amdgcn_s_wait_tensorcnt(0);
    }
    __syncthreads();
    layer32(bufA, 512, bufB, 128, W0, b0, 512, 128, true, wave, lane); __syncthreads();
    layer32(bufB, 128, bufA, 256, W1, b1, 128, 256, true, wave, lane); __syncthreads();

    // final 256 -> 607 via padded 256x608 weights; only the store is guarded
    for (int t = wave; t < 38; t += 8) {
        v8f acc0 = {}, acc1 = {};
        wmma_tile_pair(bufA, 256, W2p, 608, 256, t * 16, lane, acc0, acc1);
        const int n = t * 16 + l;
        if (n < OUTc) {
            const float bn = b2p[n];
#pragma unroll
            for (int v = 0; v < 8; ++v) {
                const int m = v + half * 8;
                outObj[(size_t)(r0 + m     ) * OUTc + n] = acc0[v] + bn;
                outObj[(size_t)(r0 + m + 16) * OUTc + n] = acc1[v] + bn;
            }
        }
    }
}

// ---------------------------------------------------------------------------
// Relation network: one WG per (i,j) pair, 32 rows = full batch in one pass.
// pn rows are pooled ReLU outputs (>= 0), so relu(concat(fi,fj)) == the rows
// themselves -> stage uI/uJ with two contiguous 64KB TDM loads and run layer 1
// as two K-half accumulations against r1_W[0:512] and r1_W[512:1024].
// ---------------------------------------------------------------------------
__global__ void __launch_bounds__(256) relation_kernel(
    const float* __restrict__ pn,
    const float* __restrict__ r1W, const float* __restrict__ r1b,
    const float* __restrict__ r2W, const float* __restrict__ r2b,
    const float* __restrict__ m1W, const float* __restrict__ m1b,
    const float* __restrict__ m2W, const float* __restrict__ m2b,
    float* __restrict__ relOut, float* __restrict__ multiOut)
{
    extern __shared__ float smem[];
    float* uI   = smem;                 // 32*512
    float* uJ   = smem + 32 * 512;      // 32*512
    float* actA = smem + 64 * 512;      // 32*512
    float* actB = smem + 96 * 512;      // 32*512

    const int pair = blockIdx.x;
    const int i = pair / Nc, j = pair % Nc;
    const int tid = threadIdx.x, lane = tid & 31, wave = tid >> 5;
    const int half = lane >> 4, l = lane & 15;

    if (wave == 0) {                    // two async 64KB stages via TDM
        tdm_load_lds(0u,                  pn + (size_t)i * Bc * Dc, 32 * 512);
        tdm_load_lds(32u * 512u * 4u,     pn + (size_t)j * Bc * Dc, 32 * 512);
        __builtin_amdgcn_s_wait_tensorcnt(0);
    }
    __syncthreads();

    // layer 1: 1024 -> 512 as two 512-K accumulations
    for (int t = wave; t < 32; t += 8) {
        v8f acc0 = {}, acc1 = {};
        wmma_tile_pair(uI, 512, r1W,             512, 512, t * 16, lane, acc0, acc1);
        wmma_tile_pair(uJ, 512, r1W + 512 * 512, 512, 512, t * 16, lane, acc0, acc1);
        epi_pair_lds(acc0, acc1, actA, 512, r1b, t * 16, lane, true);
    }
    __syncthreads();

    // layer 2 -> "rel" output (pre-ReLU stored, ReLU kept for next layer)
    const size_t pairBase = ((size_t)i * Nc + j) * Bc;
    for (int t = wave; t < 32; t += 8) {
        v8f acc0 = {}, acc1 = {};
        wmma_tile_pair(actA, 512, r2W, 512, 512, t * 16, lane, acc0, acc1);
        const int n = t * 16 + l;
        const float bn = r2b[n];
#pragma unroll
        for (int v = 0; v < 8; ++v) {
            const int m = v + half * 8;
            const float v0 = acc0[v] + bn;
            const float v1 = acc1[v] + bn;
            relOut[(pairBase + m     ) * Dc + n] = v0;
            relOut[(pairBase + m + 16) * Dc + n] = v1;
            actB[(m     ) * 512 + n] = fmaxf(v0, 0.f);
            actB[(m + 16) * 512 + n] = fmaxf(v1, 0.f);
        }
    }
    __syncthreads();
    layer32(actB, 512, actA, 512, m1W, m1b, 512, 512, true, wave, lane);
    __syncthreads();
    // layer 4 -> "multi" output
    for (int t = wave; t < 32; t += 8) {
        v8f acc0 = {}, acc1 = {};
        wmma_tile_pair(actA, 512, m2W, 512, 512, t * 16, lane, acc0, acc1);
        const int n = t * 16 + l;
        const float bn = m2b[n];
#pragma unroll
        for (int v = 0; v < 8; ++v) {
            const int m = v + half * 8;
            multiOut[(pairBase + m     ) * Dc + n] = acc0[v] + bn;
            multiOut[(pairBase + m + 16) * Dc + n] = acc1[v] + bn;
        }
    }
}

// ---------------------------------------------------------------------------
extern "C" void kernel_launch(void* const* d_in, const int* in_sizes, int n_in,
                              void* d_out, int out_size, void* d_ws, size_t ws_size,
                              hipStream_t stream)
{
    (void)in_sizes; (void)n_in; (void)out_size; (void)ws_size;
    const float* objects = (const float*)d_in[1];
    const float* deW0 = (const float*)d_in[3];  const float* deb0 = (const float*)d_in[4];
    const float* deW1 = (const float*)d_in[5];  const float* deb1 = (const float*)d_in[6];
    const float* deW2 = (const float*)d_in[7];  const float* deb2 = (const float*)d_in[8];
    const float* deW3 = (const float*)d_in[9];  const float* deb3 = (const float*)d_in[10];
    const float* deW4 = (const float*)d_in[11]; const float* deb4 = (const float*)d_in[12];
    const float* seW0 = (const float*)d_in[13]; const float* seb0 = (const float*)d_in[14];
    const float* seW1 = (const float*)d_in[15]; const float* seb1 = (const float*)d_in[16];
    const float* seW2 = (const float*)d_in[17]; const float* seb2 = (const float*)d_in[18];
    const float* omW0 = (const float*)d_in[19]; const float* omb0 = (const float*)d_in[20];
    const float* omW1 = (const float*)d_in[21]; const float* omb1 = (const float*)d_in[22];
    const float* omW2 = (const float*)d_in[23]; const float* omb2 = (const float*)d_in[24];
    const float* r1W = (const float*)d_in[25];  const float* r1b = (const float*)d_in[26];
    const float* r2W = (const float*)d_in[27];  const float* r2b = (const float*)d_in[28];
    const float* m1W = (const float*)d_in[29];  const float* m1b = (const float*)d_in[30];
    const float* m2W = (const float*)d_in[31];  const float* m2b = (const float*)d_in[32];

    // workspace layout
    float* objFeat = (float*)d_ws;                                  // [N,B,512]
    float* pnObj   = objFeat + (size_t)Nc * Bc * Dc;                // [N,B,512]
    float* deW0p   = pnObj + (size_t)Nc * Bc * Dc;                  // 8x64
    float* seW0p   = deW0p + 8 * 64;                                // 8x32
    float* omW2p   = seW0p + 8 * 32;                                // 256x608
    float* omb2p   = omW2p + 256 * 608;                             // 608

    float* outObj   = (float*)d_out;                                // [N,B,607]
    float* relOut   = outObj + (size_t)Nc * Bc * OUTc;              // [N,N,B,512]
    float* multiOut = relOut + (size_t)Nc * Nc * Bc * Dc;           // [N,N,B,512]

    // zero pooled-feature accumulators (ReLU outputs >= 0, so 0 acts as -inf)
    hipMemsetAsync(d_ws, 0, (size_t)2 * Nc * Bc * Dc * sizeof(float), stream);

    prep_weights_kernel<<<64, 256, 0, stream>>>(deW0, seW0, omW2, omb2,
                                                deW0p, seW0p, omW2p, omb2p);

    const int encWGs = Bc * Nc * (Pc / 128);                        // 13312
    const size_t encLds = (2 * 32 * 512 + 512) * sizeof(float);     // 130 KB
    dense_encoder_kernel<<<encWGs, 256, encLds, stream>>>(
        objects, deW0p, deb0, deW1, deb1, deW2, deb2, deW3, deb3, deW4, deb4, objFeat);
    sparse_encoder_kernel<<<encWGs, 256, encLds, stream>>>(
        objects, seW0p, seb0, seW1, seb1, seW2, seb2, pnObj);

    const size_t omLds = (32 * 512 + 32 * 128) * sizeof(float);     // 80 KB
    object_mlp_kernel<<<(Nc * Bc) / 32, 256, omLds, stream>>>(
        objFeat, omW0, omb0, omW1, omb1, omW2p, omb2p, outObj);

    const size_t relLds = (4 * 32 * 512) * sizeof(float);           // 256 KB
    relation_kernel<<<Nc * Nc, 256, relLds, stream>>>(
        pnObj, r1W, r1b, r2W, r2b, m1W, m1b, m2W, m2b, relOut, multiOut);
}